// MolecularGNN_71734543777946
// MI455X (gfx1250) — compile-verified
//
#include <hip/hip_runtime.h>
#include <hip/hip_bf16.h>

#define NNODE 50000
#define NEDGE 600000
#define EF    (NEDGE + NNODE)
#define ATOM  64
#define BOND  16
#define HID   128
#define HEADS 4
#define CH    32
#define NLAYER 3

typedef float v2f __attribute__((ext_vector_type(2)));
typedef float v8f __attribute__((ext_vector_type(8)));

// ---------- monotonic float<->uint mapping for atomicMax on floats ----------
__device__ __forceinline__ unsigned fflip(float f) {
  unsigned u = __float_as_uint(f);
  return (u & 0x80000000u) ? ~u : (u | 0x80000000u);
}
__device__ __forceinline__ float funflip(unsigned u) {
  return __uint_as_float((u & 0x80000000u) ? (u ^ 0x80000000u) : ~u);
}

// ---------------------------------------------------------------------------
// Dense GEMM  C[M x 128] = A[M x K] @ W[K x 128] (+bias), fp32 WMMA 16x16x4.
// Block: 8 wave32s; block tile 64 rows x 128 cols; wave tile 16 x 64
// (four 16x16 accumulators).
//  - A tile (64 x K) staged into LDS with GLOBAL_LOAD_ASYNC_TO_LDS_B128
//    (ASYNCcnt path), W staged transposed so every WMMA fragment is one
//    contiguous 8B-aligned ds_load_b64.
//  - A frag: lane(lo,hi) holds A[r0+lo][k0+2*hi + {0,1}]   (16x4 fp32 layout)
//  - B frag: lane(lo,hi) holds W[k0+2*hi + {0,1}][c0+lo]
//  - D:      lane(lo,hi) vgpr v holds C[r0 + v + 8*hi][c0+lo]
// LDS: K=128 -> 96KB, K=64 -> 48KB (WGP pool is 320KB).
// ---------------------------------------------------------------------------
template<int K>
__global__ __launch_bounds__(256)
void gemm_n128_kernel(const float* __restrict__ A, const float* __restrict__ W,
                      const float* __restrict__ bias, float* __restrict__ C, int M) {
  __shared__ float sA[64 * K];     // block A tile, row-major
  __shared__ float sWT[HID * K];   // W transposed: sWT[c*K + k]

  const long long rowbase = (long long)blockIdx.x * 64;

  // ---- async-stage A tile: one b128 (4 floats) per issue ----------------
  for (int t = threadIdx.x; t < 64 * (K / 4); t += 256) {
    const int r  = t / (K / 4);
    const int cc = (t % (K / 4)) * 4;
    long long rr = rowbase + r;
    if (rr >= M) rr = M - 1;                       // clamp; stores are guarded
    unsigned long long src = (unsigned long long)(const void*)(A + rr * K + cc);
    unsigned dstLds = (unsigned)(unsigned long long)(const void*)&sA[r * K + cc];
    asm volatile("global_load_async_to_lds_b128 %0, %1, off"
                 :: "v"(dstLds), "v"(src) : "memory");
  }
  // ---- stage W transposed through the regular path ----------------------
  for (int i = threadIdx.x; i < K * HID; i += 256) {
    const int k = i / HID, c = i % HID;
    sWT[c * K + k] = W[i];
  }
  asm volatile("s_wait_asynccnt 0x0" ::: "memory");
  __syncthreads();

  const int lane = threadIdx.x & 31;
  const int wave = threadIdx.x >> 5;
  const int lo = lane & 15;
  const int hi = lane >> 4;
  const int rloc = (wave >> 1) * 16 + lo;          // local row in tile
  const int c0 = (wave & 1) * 64;

  v8f acc0 = {}, acc1 = {}, acc2 = {}, acc3 = {};
  for (int k0 = 0; k0 < K; k0 += 4) {
    const int ka = k0 + 2 * hi;
    v2f a  = *(const v2f*)&sA[rloc * K + ka];
    v2f b0 = *(const v2f*)&sWT[(c0      + lo) * K + ka];
    v2f b1 = *(const v2f*)&sWT[(c0 + 16 + lo) * K + ka];
    v2f b2 = *(const v2f*)&sWT[(c0 + 32 + lo) * K + ka];
    v2f b3 = *(const v2f*)&sWT[(c0 + 48 + lo) * K + ka];
    acc0 = __builtin_amdgcn_wmma_f32_16x16x4_f32(false, a, false, b0, (short)0, acc0, false, false);
    acc1 = __builtin_amdgcn_wmma_f32_16x16x4_f32(false, a, false, b1, (short)0, acc1, false, false);
    acc2 = __builtin_amdgcn_wmma_f32_16x16x4_f32(false, a, false, b2, (short)0, acc2, false, false);
    acc3 = __builtin_amdgcn_wmma_f32_16x16x4_f32(false, a, false, b3, (short)0, acc3, false, false);
  }

  const int r0 = (int)rowbase + (wave >> 1) * 16;
  #pragma unroll
  for (int t = 0; t < 4; ++t) {
    v8f acc = (t == 0) ? acc0 : (t == 1) ? acc1 : (t == 2) ? acc2 : acc3;
    const int col = c0 + t * 16 + lo;
    const float bv = bias ? bias[col] : 0.0f;
    #pragma unroll
    for (int v = 0; v < 8; ++v) {
      const int row = r0 + v + 8 * hi;
      if (row < M) C[(long long)row * HID + col] = acc[v] + bv;
    }
  }
}

// ---------------------------------------------------------------------------
// w2[l][k][h] = sum_c lin[l][k][h*CH+c] * att[l][h][c]   (collapse att vecs)
// ---------------------------------------------------------------------------
__global__ void prep_w2_kernel(const float* __restrict__ lin, const float* __restrict__ att,
                               float* __restrict__ w2) {
  int t = blockIdx.x * blockDim.x + threadIdx.x;
  if (t >= NLAYER * HID * HEADS) return;
  int l = t / (HID * HEADS);
  int k = (t / HEADS) % HID;
  int h = t % HEADS;
  const float* wl = lin + ((long long)l * HID + k) * (HEADS * CH) + h * CH;
  const float* al = att + (l * HEADS + h) * CH;
  float s = 0.f;
  #pragma unroll
  for (int c = 0; c < CH; ++c) s += wl[c] * al[c];
  w2[t] = s;
}

__global__ void zero_f32_kernel(float* __restrict__ p, long long n) {
  long long i = (long long)blockIdx.x * blockDim.x + threadIdx.x;
  if (i < n) p[i] = 0.f;
}

// ---------------------------------------------------------------------------
// One wave32 per edge: bond embedding (16->128) computed on the fly,
// accumulated into loop_sum[dst] + deg[dst], and fused a_edge for ALL layers.
// ---------------------------------------------------------------------------
__global__ __launch_bounds__(256)
void edge_init_kernel(const int* __restrict__ ei, const float* __restrict__ eattr,
                      const float* __restrict__ bond_w, const float* __restrict__ bond_b,
                      const float* __restrict__ w_edge2,
                      float* __restrict__ loop_sum, float* __restrict__ deg,
                      float* __restrict__ a_edge) {
  const long long gw = ((long long)blockIdx.x * blockDim.x + threadIdx.x) >> 5;
  const int lane = threadIdx.x & 31;
  if (gw >= NEDGE) return;
  const int e = (int)gw;
  const int dst = ei[NEDGE + e];

  float attr[BOND];
  #pragma unroll
  for (int j = 0; j < BOND; ++j) attr[j] = eattr[(long long)e * BOND + j];

  float part[NLAYER * HEADS];
  #pragma unroll
  for (int t = 0; t < NLAYER * HEADS; ++t) part[t] = 0.f;

  #pragma unroll
  for (int i = 0; i < 4; ++i) {
    const int c = i * 32 + lane;
    float s = bond_b[c];
    #pragma unroll
    for (int j = 0; j < BOND; ++j) s += attr[j] * bond_w[j * HID + c];
    atomicAdd(&loop_sum[(long long)dst * HID + c], s);
    #pragma unroll
    for (int l = 0; l < NLAYER; ++l)
      #pragma unroll
      for (int h = 0; h < HEADS; ++h)
        part[l * HEADS + h] += s * w_edge2[(l * HID + c) * HEADS + h];
  }
  #pragma unroll
  for (int t = 0; t < NLAYER * HEADS; ++t)
    for (int m = 16; m >= 1; m >>= 1) part[t] += __shfl_xor(part[t], m, 32);
  if (lane == 0) {
    #pragma unroll
    for (int t = 0; t < NLAYER * HEADS; ++t) a_edge[(long long)e * 12 + t] = part[t];
    atomicAdd(&deg[dst], 1.f);
  }
}

// One wave32 per node: loop_attr = loop_sum/max(deg,1); a_edge for self-loops.
__global__ __launch_bounds__(256)
void loop_edge_kernel(const float* __restrict__ w_edge2, float* __restrict__ loop_attr,
                      const float* __restrict__ deg, float* __restrict__ a_edge) {
  const long long gw = ((long long)blockIdx.x * blockDim.x + threadIdx.x) >> 5;
  const int lane = threadIdx.x & 31;
  if (gw >= NNODE) return;
  const int n = (int)gw;
  const float inv = 1.f / fmaxf(deg[n], 1.f);

  float part[NLAYER * HEADS];
  #pragma unroll
  for (int t = 0; t < NLAYER * HEADS; ++t) part[t] = 0.f;

  #pragma unroll
  for (int i = 0; i < 4; ++i) {
    const int c = i * 32 + lane;
    float la = loop_attr[(long long)n * HID + c] * inv;
    loop_attr[(long long)n * HID + c] = la;
    #pragma unroll
    for (int l = 0; l < NLAYER; ++l)
      #pragma unroll
      for (int h = 0; h < HEADS; ++h)
        part[l * HEADS + h] += la * w_edge2[(l * HID + c) * HEADS + h];
  }
  #pragma unroll
  for (int t = 0; t < NLAYER * HEADS; ++t)
    for (int m = 16; m >= 1; m >>= 1) part[t] += __shfl_xor(part[t], m, 32);
  if (lane == 0) {
    #pragma unroll
    for (int t = 0; t < NLAYER * HEADS; ++t) a_edge[(long long)(NEDGE + n) * 12 + t] = part[t];
  }
}

// a_src/a_dst = h @ w_src2[l] / w_dst2[l]   (thread per (node, head))
__global__ void node_att_kernel(const float* __restrict__ h, const float* __restrict__ w_src2,
                                const float* __restrict__ w_dst2, int l,
                                float* __restrict__ a_src, float* __restrict__ a_dst) {
  int t = blockIdx.x * blockDim.x + threadIdx.x;
  if (t >= NNODE * HEADS) return;
  const int n = t >> 2, hh = t & 3;
  const float* ws = w_src2 + (long long)l * HID * HEADS;
  const float* wd = w_dst2 + (long long)l * HID * HEADS;
  const float* hr = h + (long long)n * HID;
  float s = 0.f, d = 0.f;
  for (int k = 0; k < HID; ++k) {
    float hv = hr[k];
    s += hv * ws[k * HEADS + hh];
    d += hv * wd[k * HEADS + hh];
  }
  a_src[t] = s; a_dst[t] = d;
}

__global__ void init_layer_kernel(float* __restrict__ agg, float* __restrict__ denom,
                                  unsigned* __restrict__ amax) {
  long long i = (long long)blockIdx.x * blockDim.x + threadIdx.x;
  if (i < (long long)NNODE * HID) agg[i] = 0.f;
  if (i < NNODE * HEADS) { denom[i] = 0.f; amax[i] = 0u; }   // 0 == fflip(-huge)
}

__global__ void att_max_kernel(const int* __restrict__ ei, const float* __restrict__ a_src,
                               const float* __restrict__ a_dst, const float* __restrict__ a_edge,
                               int l, float* __restrict__ alpha, unsigned* __restrict__ amax) {
  long long t = (long long)blockIdx.x * blockDim.x + threadIdx.x;
  if (t >= (long long)EF * HEADS) return;
  const long long e = t >> 2; const int hh = (int)(t & 3);
  int s, d;
  if (e < NEDGE) { s = ei[e]; d = ei[NEDGE + e]; } else { s = d = (int)(e - NEDGE); }
  float al = a_src[s * HEADS + hh] + a_dst[d * HEADS + hh] + a_edge[e * 12 + l * HEADS + hh];
  al = (al > 0.f) ? al : 0.2f * al;           // leaky_relu(0.2)
  alpha[t] = al;
  atomicMax(&amax[d * HEADS + hh], fflip(al));
}

__global__ void att_exp_kernel(const int* __restrict__ ei, float* __restrict__ alpha,
                               const unsigned* __restrict__ amax, float* __restrict__ denom) {
  long long t = (long long)blockIdx.x * blockDim.x + threadIdx.x;
  if (t >= (long long)EF * HEADS) return;
  const long long e = t >> 2; const int hh = (int)(t & 3);
  int d;
  if (e < NEDGE) { d = ei[NEDGE + e]; } else { d = (int)(e - NEDGE); }
  float ez = __expf(alpha[t] - funflip(amax[d * HEADS + hh]));
  alpha[t] = ez;
  atomicAdd(&denom[d * HEADS + hh], ez);
}

// One wave32 per edge: weighted message scatter (lane = channel % 32, head = channel/32)
__global__ __launch_bounds__(256)
void scatter_kernel(const int* __restrict__ ei, const float* __restrict__ alpha,
                    const float* __restrict__ denom, const float* __restrict__ xw,
                    float* __restrict__ agg) {
  const long long gw = ((long long)blockIdx.x * blockDim.x + threadIdx.x) >> 5;
  const int lane = threadIdx.x & 31;
  if (gw >= EF) return;
  const long long e = gw;
  int s, d;
  if (e < NEDGE) { s = ei[e]; d = ei[NEDGE + e]; } else { s = d = (int)(e - NEDGE); }
  float wv[HEADS];
  #pragma unroll
  for (int hh = 0; hh < HEADS; ++hh)
    wv[hh] = alpha[e * HEADS + hh] / (denom[d * HEADS + hh] + 1e-16f);
  #pragma unroll
  for (int i = 0; i < 4; ++i) {
    const int c = i * CH + lane;
    atomicAdd(&agg[(long long)d * HID + c], xw[(long long)s * HID + c] * wv[i]);
  }
}

// Per-channel batch statistics (block per channel)
__global__ __launch_bounds__(256)
void bn_stats_kernel(const float* __restrict__ agg, float* __restrict__ mu, float* __restrict__ var) {
  __shared__ float ss[256], ss2[256];
  const int c = blockIdx.x;
  float s = 0.f, s2 = 0.f;
  for (int n = threadIdx.x; n < NNODE; n += 256) {
    float v = agg[(long long)n * HID + c];
    s += v; s2 += v * v;
  }
  ss[threadIdx.x] = s; ss2[threadIdx.x] = s2;
  __syncthreads();
  for (int o = 128; o > 0; o >>= 1) {
    if (threadIdx.x < o) { ss[threadIdx.x] += ss[threadIdx.x + o]; ss2[threadIdx.x] += ss2[threadIdx.x + o]; }
    __syncthreads();
  }
  if (threadIdx.x == 0) {
    float m = ss[0] / (float)NNODE;
    mu[c] = m;
    var[c] = ss2[0] / (float)NNODE - m * m;
  }
}

// BN + ReLU + residual. (conv_bias cancels inside training-mode BN.)
__global__ void bn_apply_kernel(const float* __restrict__ agg, const float* __restrict__ mu,
                                const float* __restrict__ var, const float* __restrict__ gamma,
                                const float* __restrict__ beta, int l, float* __restrict__ h) {
  long long i = (long long)blockIdx.x * blockDim.x + threadIdx.x;
  if (i >= (long long)NNODE * HID) return;
  const int c = (int)(i & (HID - 1));
  float xn = (agg[i] - mu[c]) * rsqrtf(var[c] + 1e-5f) * gamma[l * HID + c] + beta[l * HID + c];
  h[i] += fmaxf(xn, 0.f);
}

// ---------------------------------------------------------------------------
extern "C" void kernel_launch(void* const* d_in, const int* in_sizes, int n_in,
                              void* d_out, int out_size, void* d_ws, size_t ws_size,
                              hipStream_t stream) {
  const float* x        = (const float*)d_in[0];
  const int*   ei       = (const int*)d_in[1];
  const float* eattr    = (const float*)d_in[2];
  const float* atom_w   = (const float*)d_in[3];
  const float* atom_b   = (const float*)d_in[4];
  const float* bond_w   = (const float*)d_in[5];
  const float* bond_b   = (const float*)d_in[6];
  const float* lin_w    = (const float*)d_in[7];
  const float* lin_ew   = (const float*)d_in[8];
  const float* att_src  = (const float*)d_in[9];
  const float* att_dst  = (const float*)d_in[10];
  const float* att_edge = (const float*)d_in[11];
  // d_in[12] conv_bias: per-channel constant before training-mode BN -> cancels
  const float* bn_gamma = (const float*)d_in[13];
  const float* bn_beta  = (const float*)d_in[14];
  const float* out_w    = (const float*)d_in[15];
  const float* out_b    = (const float*)d_in[16];
  float* out = (float*)d_out;
  (void)in_sizes; (void)n_in; (void)out_size; (void)ws_size;

  float* p = (float*)d_ws;
  float* h       = p; p += (long long)NNODE * HID;
  float* xw      = p; p += (long long)NNODE * HID;
  float* agg     = p; p += (long long)NNODE * HID;
  float* loopb   = p; p += (long long)NNODE * HID;   // loop_sum -> loop_attr
  float* deg     = p; p += NNODE;
  float* a_edge  = p; p += (long long)EF * 12;
  float* alpha   = p; p += (long long)EF * HEADS;
  float* a_src   = p; p += NNODE * HEADS;
  float* a_dst   = p; p += NNODE * HEADS;
  float* denom   = p; p += NNODE * HEADS;
  unsigned* amax = (unsigned*)p; p += NNODE * HEADS;
  float* w_src2  = p; p += NLAYER * HID * HEADS;
  float* w_dst2  = p; p += NLAYER * HID * HEADS;
  float* w_edge2 = p; p += NLAYER * HID * HEADS;
  float* mu      = p; p += HID;
  float* var     = p; p += HID;

  const int T = 256;
  const int w2n = NLAYER * HID * HEADS;

  prep_w2_kernel<<<(w2n + T - 1) / T, T, 0, stream>>>(lin_w,  att_src,  w_src2);
  prep_w2_kernel<<<(w2n + T - 1) / T, T, 0, stream>>>(lin_w,  att_dst,  w_dst2);
  prep_w2_kernel<<<(w2n + T - 1) / T, T, 0, stream>>>(lin_ew, att_edge, w_edge2);

  gemm_n128_kernel<ATOM><<<(NNODE + 63) / 64, T, 0, stream>>>(x, atom_w, atom_b, h, NNODE);

  zero_f32_kernel<<<(int)(((long long)NNODE * HID + T - 1) / T), T, 0, stream>>>(loopb, (long long)NNODE * HID);
  zero_f32_kernel<<<(NNODE + T - 1) / T, T, 0, stream>>>(deg, NNODE);

  edge_init_kernel<<<(int)(((long long)NEDGE * 32 + T - 1) / T), T, 0, stream>>>(
      ei, eattr, bond_w, bond_b, w_edge2, loopb, deg, a_edge);
  loop_edge_kernel<<<(int)(((long long)NNODE * 32 + T - 1) / T), T, 0, stream>>>(
      w_edge2, loopb, deg, a_edge);

  for (int l = 0; l < NLAYER; ++l) {
    gemm_n128_kernel<HID><<<(NNODE + 63) / 64, T, 0, stream>>>(
        h, lin_w + (long long)l * HID * HID, nullptr, xw, NNODE);
    node_att_kernel<<<(NNODE * HEADS + T - 1) / T, T, 0, stream>>>(h, w_src2, w_dst2, l, a_src, a_dst);
    init_layer_kernel<<<(int)(((long long)NNODE * HID + T - 1) / T), T, 0, stream>>>(agg, denom, amax);
    att_max_kernel<<<(int)(((long long)EF * HEADS + T - 1) / T), T, 0, stream>>>(
        ei, a_src, a_dst, a_edge, l, alpha, amax);
    att_exp_kernel<<<(int)(((long long)EF * HEADS + T - 1) / T), T, 0, stream>>>(ei, alpha, amax, denom);
    scatter_kernel<<<(int)(((long long)EF * 32 + T - 1) / T), T, 0, stream>>>(ei, alpha, denom, xw, agg);
    bn_stats_kernel<<<HID, T, 0, stream>>>(agg, mu, var);
    bn_apply_kernel<<<(int)(((long long)NNODE * HID + T - 1) / T), T, 0, stream>>>(
        agg, mu, var, bn_gamma, bn_beta, l, h);
  }

  gemm_n128_kernel<HID><<<(NNODE + 63) / 64, T, 0, stream>>>(h, out_w, out_b, out, NNODE);
}